// BatchBasedSemiHardMarginLoss_10333691314643
// MI455X (gfx1250) — compile-verified
//
#include <hip/hip_runtime.h>
#include <hip/hip_bf16.h>

// ---------------------------------------------------------------------------
// Problem constants (B=8192, D=256 from the reference setup)
// ---------------------------------------------------------------------------
#define BDIM   8192
#define DDIM   256
#define MARGIN 0.2f
#define NTILES (BDIM / 16)     // 512 column tiles of 16
#define NQUADS (NTILES / 4)    // 128 tile quads (4-way N blocking)

typedef float v2f __attribute__((ext_vector_type(2)));
typedef float v8f __attribute__((ext_vector_type(8)));

// ---------------------------------------------------------------------------
// Async global -> LDS staging of one 16-row A block (16 KB), using the CDNA5
// Tensor/Async path: GLOBAL_LOAD_ASYNC_TO_LDS_B128 (ASYNCcnt tracked).
// ldsPtr low 32 bits of the generic (flat) address = addrspace(3) byte offset.
// Каждый thread copies 4 x 16B. Caller must __syncthreads() afterwards.
// ---------------------------------------------------------------------------
__device__ __forceinline__ void async_stage_16rows(const float* __restrict__ gsrc,
                                                   float* ldsDst, int tid) {
    unsigned dstOff = (unsigned)(unsigned long long)(void*)ldsDst + (unsigned)tid * 16u;
    unsigned long long gaddr = (unsigned long long)gsrc + (unsigned long long)tid * 16ull;
#pragma unroll
    for (int i = 0; i < 4; ++i) {
        asm volatile("global_load_async_to_lds_b128 %0, %1, off"
                     :: "v"(dstOff + (unsigned)(i * 4096)),
                        "v"(gaddr + (unsigned long long)(i * 4096))
                     : "memory");
    }
    asm volatile("s_wait_asynccnt 0x0" ::: "memory");
}

// ---------------------------------------------------------------------------
// JAX threefry2x32-20 (device) — exact rotation/key schedule as jax._src.prng
// ---------------------------------------------------------------------------
__device__ __forceinline__ unsigned d_rotl(unsigned x, int r) {
    return (x << r) | (x >> (32 - r));
}

__device__ __forceinline__ void d_threefry(unsigned k0, unsigned k1,
                                           unsigned c0, unsigned c1,
                                           unsigned& o0, unsigned& o1) {
    unsigned ks[3] = {k0, k1, k0 ^ k1 ^ 0x1BD11BDAu};
    unsigned x0 = c0 + ks[0];
    unsigned x1 = c1 + ks[1];
    const int rotA[4] = {13, 15, 26, 6};
    const int rotB[4] = {17, 29, 16, 24};
#pragma unroll
    for (int g = 0; g < 5; ++g) {
        const int* R = (g & 1) ? rotB : rotA;
#pragma unroll
        for (int r = 0; r < 4; ++r) {
            x0 += x1;
            x1 = d_rotl(x1, R[r]);
            x1 ^= x0;
        }
        x0 += ks[(g + 1) % 3];
        x1 += ks[(g + 2) % 3] + (unsigned)(g + 1);
    }
    o0 = x0; o1 = x1;
}

// Gumbel sample for flat element t of a (8192, 8192) draw, matching
// jax.random.gumbel(key, (B,B), float32): counts = iota split in halves.
__device__ __forceinline__ float d_gumbel(unsigned k0, unsigned k1, unsigned t) {
    const unsigned HALF = 1u << 25;  // (8192*8192)/2
    unsigned c0, c1, o0, o1;
    if (t < HALF) { c0 = t;        c1 = t + HALF; }
    else          { c0 = t - HALF; c1 = t;        }
    d_threefry(k0, k1, c0, c1, o0, o1);
    unsigned bits = (t < HALF) ? o0 : o1;
    unsigned fb = (bits >> 9) | 0x3F800000u;
    float f = __uint_as_float(fb) - 1.0f;          // [0,1)
    const float tiny = 1.1754943508222875e-38f;    // finfo(f32).tiny
    float u = fmaxf(f * (1.0f - tiny) + tiny, tiny);
    return -__logf(-__logf(u));
}

// ---------------------------------------------------------------------------
// Kernel 1: semihard mining. Workgroup = 16 rows of sim (A rows in LDS via
// async copy). Wave 0 first computes the diagonal 16x16 tile with WMMA so the
// pos values are bitwise-identical to the sweep; they are staged via LDS and
// written to diag[]. Then 8 waves sweep 128 tile-quads (4-way N blocking);
// Gumbel is evaluated only for in-band semihard candidates. Rows with no
// candidate are flagged for the fallback kernel.
// ---------------------------------------------------------------------------
__global__ void __launch_bounds__(256)
mine_kernel(const float* __restrict__ A, const float* __restrict__ Bm,
            float* __restrict__ diag, int* __restrict__ neg,
            int* __restrict__ found, unsigned k0, unsigned k1) {
    __shared__ __align__(16) float As[16 * DDIM];
    __shared__ float stile[16][16];
    __shared__ float rsg[8][16]; __shared__ int rsi[8][16];

    const int tid  = threadIdx.x;
    const int lane = tid & 31;
    const int wave = tid >> 5;
    const int rowBase = blockIdx.x * 16;

    const int nl = lane & 15;   // N (and A-row) index within tile
    const int hs = lane >> 4;   // K-pair selector

    async_stage_16rows(A + (size_t)rowBase * DDIM, As, tid);
    __syncthreads();

    const float* aptr = As + nl * DDIM + 2 * hs;

    // ---- Step 1: diagonal tile (wave 0 only) -> pos via identical WMMA path
    if (wave == 0) {
        const float* bptr = Bm + (size_t)(rowBase + nl) * DDIM + 2 * hs;
        v8f acc = {0.f, 0.f, 0.f, 0.f, 0.f, 0.f, 0.f, 0.f};
#pragma unroll 4
        for (int k = 0; k < DDIM; k += 4) {
            v2f a = *(const v2f*)(aptr + k);
            v2f b = *(const v2f*)(bptr + k);
            acc = __builtin_amdgcn_wmma_f32_16x16x4_f32(
                false, a, false, b, (short)0, acc, false, false);
        }
#pragma unroll
        for (int r = 0; r < 8; ++r) stile[r + 8 * hs][nl] = acc[r];
    }
    __syncthreads();
    if (tid < 16) diag[rowBase + tid] = stile[tid][tid];

    float pd[8]; int gi[8];
#pragma unroll
    for (int r = 0; r < 8; ++r) {
        gi[r] = rowBase + r + 8 * hs;
        pd[r] = stile[r + 8 * hs][r + 8 * hs];
    }

    // ---- Step 2: sweep 128 tile quads (4-way N blocking)
    float sg[8]; int si[8];
#pragma unroll
    for (int r = 0; r < 8; ++r) { sg[r] = -__builtin_inff(); si[r] = 0; }

    for (int q = wave; q < NQUADS; q += 8) {
        const int colBase = q * 64;
        const float* bptr0 = Bm + (size_t)(colBase +  0 + nl) * DDIM + 2 * hs;
        const float* bptr1 = Bm + (size_t)(colBase + 16 + nl) * DDIM + 2 * hs;
        const float* bptr2 = Bm + (size_t)(colBase + 32 + nl) * DDIM + 2 * hs;
        const float* bptr3 = Bm + (size_t)(colBase + 48 + nl) * DDIM + 2 * hs;
        // Speculative prefetch of the next quad this wave will touch.
        if (q + 8 < NQUADS) {
            const int nb = (q + 8) * 64;
            __builtin_prefetch(Bm + (size_t)(nb +  0 + nl) * DDIM, 0, 3);
            __builtin_prefetch(Bm + (size_t)(nb + 16 + nl) * DDIM, 0, 3);
            __builtin_prefetch(Bm + (size_t)(nb + 32 + nl) * DDIM, 0, 3);
            __builtin_prefetch(Bm + (size_t)(nb + 48 + nl) * DDIM, 0, 3);
        }
        v8f acc0 = {0.f, 0.f, 0.f, 0.f, 0.f, 0.f, 0.f, 0.f};
        v8f acc1 = acc0, acc2 = acc0, acc3 = acc0;
#pragma unroll 4
        for (int k = 0; k < DDIM; k += 4) {
            v2f a  = *(const v2f*)(aptr + k);
            v2f b0 = *(const v2f*)(bptr0 + k);
            v2f b1 = *(const v2f*)(bptr1 + k);
            v2f b2 = *(const v2f*)(bptr2 + k);
            v2f b3 = *(const v2f*)(bptr3 + k);
            acc0 = __builtin_amdgcn_wmma_f32_16x16x4_f32(
                false, a, false, b0, (short)0, acc0, false, false);
            acc1 = __builtin_amdgcn_wmma_f32_16x16x4_f32(
                false, a, false, b1, (short)0, acc1, false, false);
            acc2 = __builtin_amdgcn_wmma_f32_16x16x4_f32(
                false, a, false, b2, (short)0, acc2, false, false);
            acc3 = __builtin_amdgcn_wmma_f32_16x16x4_f32(
                false, a, false, b3, (short)0, acc3, false, false);
        }
        const int gj0 = colBase + nl;
#pragma unroll
        for (int r = 0; r < 8; ++r) {
            const unsigned tbase = (unsigned)gi[r] * (unsigned)BDIM;
            float s;
            s = acc0[r];
            if (gj0 != gi[r] && s > pd[r] && s < pd[r] + MARGIN) {
                float g = d_gumbel(k0, k1, tbase + (unsigned)gj0);
                if (g > sg[r]) { sg[r] = g; si[r] = gj0; }
            }
            s = acc1[r];
            if (gj0 + 16 != gi[r] && s > pd[r] && s < pd[r] + MARGIN) {
                float g = d_gumbel(k0, k1, tbase + (unsigned)(gj0 + 16));
                if (g > sg[r]) { sg[r] = g; si[r] = gj0 + 16; }
            }
            s = acc2[r];
            if (gj0 + 32 != gi[r] && s > pd[r] && s < pd[r] + MARGIN) {
                float g = d_gumbel(k0, k1, tbase + (unsigned)(gj0 + 32));
                if (g > sg[r]) { sg[r] = g; si[r] = gj0 + 32; }
            }
            s = acc3[r];
            if (gj0 + 48 != gi[r] && s > pd[r] && s < pd[r] + MARGIN) {
                float g = d_gumbel(k0, k1, tbase + (unsigned)(gj0 + 48));
                if (g > sg[r]) { sg[r] = g; si[r] = gj0 + 48; }
            }
        }
    }

    // Reduce across the 16 lanes of each half-wave (same rows, different cols).
#pragma unroll
    for (int m = 8; m >= 1; m >>= 1) {
#pragma unroll
        for (int r = 0; r < 8; ++r) {
            float g2 = __shfl_xor(sg[r], m, 32); int i2 = __shfl_xor(si[r], m, 32);
            if (g2 > sg[r] || (g2 == sg[r] && i2 < si[r])) { sg[r] = g2; si[r] = i2; }
        }
    }
    if (nl == 0) {
#pragma unroll
        for (int r = 0; r < 8; ++r) {
            int m = r + 8 * hs;
            rsg[wave][m] = sg[r]; rsi[wave][m] = si[r];
        }
    }
    __syncthreads();
    if (tid < 16) {
        float bsg = -__builtin_inff(); int bsi = 0;
        for (int w = 0; w < 8; ++w) {
            float g = rsg[w][tid]; int i = rsi[w][tid];
            if (g > bsg || (g == bsg && i < bsi)) { bsg = g; bsi = i; }
        }
        const int row = rowBase + tid;
        found[row] = (bsg > -__builtin_inff()) ? 1 : 0;
        neg[row]   = bsi;   // valid when found; fallback kernel fixes the rest
    }
}

// ---------------------------------------------------------------------------
// Kernel 2: fallback — rows with NO semihard candidate take the argmax of
// Gumbel noise over all off-diagonal columns. One wave32 per row; waves whose
// row already found a semihard negative exit immediately (expected: all).
// ---------------------------------------------------------------------------
__global__ void __launch_bounds__(256)
fallback_kernel(const int* __restrict__ found, int* __restrict__ neg,
                unsigned k0, unsigned k1) {
    const int lane = threadIdx.x & 31;
    const int wave = threadIdx.x >> 5;
    const int row  = blockIdx.x * 8 + wave;
    if (found[row]) return;   // wave-uniform

    float bg = -__builtin_inff(); int bi = 0;
    for (int j = lane; j < BDIM; j += 32) {
        if (j == row) continue;
        float g = d_gumbel(k0, k1, (unsigned)row * (unsigned)BDIM + (unsigned)j);
        if (g > bg || (g == bg && j < bi)) { bg = g; bi = j; }
    }
#pragma unroll
    for (int m = 16; m >= 1; m >>= 1) {
        float g2 = __shfl_xor(bg, m, 32); int i2 = __shfl_xor(bi, m, 32);
        if (g2 > bg || (g2 == bg && i2 < bi)) { bg = g2; bi = i2; }
    }
    if (lane == 0) neg[row] = bi;
}

// ---------------------------------------------------------------------------
// Kernel 3: loss pass.  sum_{i,j} relu(dot(A[i], Bm[neg[j]]) - diag[j] + m)
// 4-way N blocking; one double partial per workgroup (fixed-order reduction).
// ---------------------------------------------------------------------------
__global__ void __launch_bounds__(256)
loss_kernel(const float* __restrict__ A, const float* __restrict__ Bm,
            const int* __restrict__ neg, const float* __restrict__ diag,
            double* __restrict__ partial) {
    __shared__ __align__(16) float As[16 * DDIM];
    __shared__ double wsum[8];

    const int tid  = threadIdx.x;
    const int lane = tid & 31;
    const int wave = tid >> 5;
    const int rowBase = blockIdx.x * 16;

    async_stage_16rows(A + (size_t)rowBase * DDIM, As, tid);
    __syncthreads();

    const int nl = lane & 15;
    const int hs = lane >> 4;
    const float* aptr = As + nl * DDIM + 2 * hs;

    double lsum = 0.0;
    for (int q = wave; q < NQUADS; q += 8) {
        const int colBase = q * 64;
        const int gj0 = colBase + nl;
        const float bias0 = MARGIN - diag[gj0];
        const float bias1 = MARGIN - diag[gj0 + 16];
        const float bias2 = MARGIN - diag[gj0 + 32];
        const float bias3 = MARGIN - diag[gj0 + 48];
        const float* bptr0 = Bm + (size_t)neg[gj0]      * DDIM + 2 * hs;
        const float* bptr1 = Bm + (size_t)neg[gj0 + 16] * DDIM + 2 * hs;
        const float* bptr2 = Bm + (size_t)neg[gj0 + 32] * DDIM + 2 * hs;
        const float* bptr3 = Bm + (size_t)neg[gj0 + 48] * DDIM + 2 * hs;
        v8f acc0 = {0.f, 0.f, 0.f, 0.f, 0.f, 0.f, 0.f, 0.f};
        v8f acc1 = acc0, acc2 = acc0, acc3 = acc0;
#pragma unroll 4
        for (int k = 0; k < DDIM; k += 4) {
            v2f a  = *(const v2f*)(aptr + k);
            v2f b0 = *(const v2f*)(bptr0 + k);
            v2f b1 = *(const v2f*)(bptr1 + k);
            v2f b2 = *(const v2f*)(bptr2 + k);
            v2f b3 = *(const v2f*)(bptr3 + k);
            acc0 = __builtin_amdgcn_wmma_f32_16x16x4_f32(
                false, a, false, b0, (short)0, acc0, false, false);
            acc1 = __builtin_amdgcn_wmma_f32_16x16x4_f32(
                false, a, false, b1, (short)0, acc1, false, false);
            acc2 = __builtin_amdgcn_wmma_f32_16x16x4_f32(
                false, a, false, b2, (short)0, acc2, false, false);
            acc3 = __builtin_amdgcn_wmma_f32_16x16x4_f32(
                false, a, false, b3, (short)0, acc3, false, false);
        }
        float t = 0.0f;
#pragma unroll
        for (int r = 0; r < 8; ++r) {
            t += fmaxf(acc0[r] + bias0, 0.0f);
            t += fmaxf(acc1[r] + bias1, 0.0f);
            t += fmaxf(acc2[r] + bias2, 0.0f);
            t += fmaxf(acc3[r] + bias3, 0.0f);
        }
        lsum += (double)t;
    }
#pragma unroll
    for (int m = 16; m >= 1; m >>= 1) lsum += __shfl_xor(lsum, m, 32);
    if (lane == 0) wsum[wave] = lsum;
    __syncthreads();
    if (tid == 0) {
        double s = 0.0;
        for (int w = 0; w < 8; ++w) s += wsum[w];
        partial[blockIdx.x] = s;
    }
}

// ---------------------------------------------------------------------------
// Kernel 4: finalize — sequential deterministic sum of partials.
// ---------------------------------------------------------------------------
__global__ void finalize_kernel(const double* __restrict__ p1,
                                const double* __restrict__ p2,
                                float* __restrict__ out) {
    if (threadIdx.x == 0 && blockIdx.x == 0) {
        double s1 = 0.0, s2 = 0.0;
        for (int i = 0; i < NTILES; ++i) { s1 += p1[i]; s2 += p2[i]; }
        const double N = (double)BDIM * (double)BDIM;
        out[0] = (float)(s1 / N) + (float)(s2 / N);
    }
}

// ---------------------------------------------------------------------------
// Host: threefry for jax.random.split(jax.random.key(42))
// ---------------------------------------------------------------------------
static inline unsigned h_rotl(unsigned x, int r) { return (x << r) | (x >> (32 - r)); }

static void h_threefry(unsigned k0, unsigned k1, unsigned c0, unsigned c1,
                       unsigned& o0, unsigned& o1) {
    unsigned ks[3] = {k0, k1, k0 ^ k1 ^ 0x1BD11BDAu};
    unsigned x0 = c0 + ks[0];
    unsigned x1 = c1 + ks[1];
    const int rotA[4] = {13, 15, 26, 6};
    const int rotB[4] = {17, 29, 16, 24};
    for (int g = 0; g < 5; ++g) {
        const int* R = (g & 1) ? rotB : rotA;
        for (int r = 0; r < 4; ++r) {
            x0 += x1; x1 = h_rotl(x1, R[r]); x1 ^= x0;
        }
        x0 += ks[(g + 1) % 3];
        x1 += ks[(g + 2) % 3] + (unsigned)(g + 1);
    }
    o0 = x0; o1 = x1;
}

extern "C" void kernel_launch(void* const* d_in, const int* in_sizes, int n_in,
                              void* d_out, int out_size, void* d_ws, size_t ws_size,
                              hipStream_t stream) {
    const float* ref = (const float*)d_in[0];
    const float* tar = (const float*)d_in[1];
    float* out = (float*)d_out;

    // Workspace layout (all written before read on every call):
    char* ws = (char*)d_ws;
    int*    neg1   = (int*)ws;                           // 8192 ints
    int*    neg2   = (int*)(ws + (size_t)BDIM * 4);      // 8192 ints
    int*    found1 = (int*)(ws + (size_t)BDIM * 8);      // 8192 ints
    int*    found2 = (int*)(ws + (size_t)BDIM * 12);     // 8192 ints
    float*  diag   = (float*)(ws + (size_t)BDIM * 16);   // 8192 floats
    double* p1     = (double*)(ws + (size_t)BDIM * 20);  // 512 doubles (8B aligned)
    double* p2     = p1 + NTILES;                        // 512 doubles

    // key(42) -> data (0, 42); split -> bits = threefry(key, iota(4)) halved.
    unsigned b0, b1, b2, b3;
    h_threefry(0u, 42u, 0u, 2u, b0, b2);  // counts (0,2): out0 -> bits[0], out1 -> bits[2]
    h_threefry(0u, 42u, 1u, 3u, b1, b3);  // counts (1,3): out0 -> bits[1], out1 -> bits[3]
    const unsigned k1a = b0, k1b = b1;    // first subkey  (for sim)
    const unsigned k2a = b2, k2b = b3;    // second subkey (for sim.T)

    mine_kernel<<<NTILES, 256, 0, stream>>>(ref, tar, diag, neg1, found1, k1a, k1b);
    fallback_kernel<<<BDIM / 8, 256, 0, stream>>>(found1, neg1, k1a, k1b);
    mine_kernel<<<NTILES, 256, 0, stream>>>(tar, ref, diag, neg2, found2, k2a, k2b);
    fallback_kernel<<<BDIM / 8, 256, 0, stream>>>(found2, neg2, k2a, k2b);
    loss_kernel<<<NTILES, 256, 0, stream>>>(ref, tar, neg1, diag, p1);
    loss_kernel<<<NTILES, 256, 0, stream>>>(tar, ref, neg2, diag, p2);
    finalize_kernel<<<1, 32, 0, stream>>>(p1, p2, out);
}